// RepulsionNLH_19310172963097
// MI455X (gfx1250) — compile-verified
//
#include <hip/hip_runtime.h>

#define ZMAX 92
#define TAB 8649            // (ZMAX+1)^2
#define TAB3 (TAB * 3)      // 25947 floats per table
#define LDS_FLOATS (TAB * 6) // interleaved c/a: 51894 floats = 207576 B
#define LOG2E 1.44269504088896340736f
#define PREF (0.5f * 0.52917721f)   // 0.5 * BOHR * ENERGY_UNIT

#ifndef __has_builtin
#define __has_builtin(x) 0
#endif

#if __has_builtin(__builtin_amdgcn_global_load_async_to_lds_b32)
#define HAVE_ASYNC_COPY 1
#else
#define HAVE_ASYNC_COPY 0
#endif

// async builtin takes int-typed AS(1)/AS(3) pointers (AS(1) prints as
// "__device__" in HIP diagnostics).
typedef __attribute__((address_space(1))) int g_i32;
typedef __attribute__((address_space(3))) int l_i32;

__device__ __forceinline__ void do_edge(int s, int d, float r, float w,
                                        const int* __restrict__ species,
                                        const float* __restrict__ shTab,
                                        float* __restrict__ out) {
    int zs = species[s];
    int zd = species[d];
    const float* t = shTab + 6 * (zs + ZMAX * zd);
    float m = -r * LOG2E;
    // phi = c0*exp(-a0*r) + c1*exp(-a1*r) + c2*exp(-a2*r); exp via v_exp_f32
    float phi = t[0] * __builtin_amdgcn_exp2f(t[1] * m)
              + t[2] * __builtin_amdgcn_exp2f(t[3] * m)
              + t[4] * __builtin_amdgcn_exp2f(t[5] * m);
    // Z = where(species>0, species, 0) == (float)species since species in [0,92]
    float val = PREF * (float)(zs * zd) * phi * w * __builtin_amdgcn_rcpf(r);
    unsafeAtomicAdd(out + s, val);   // global_atomic_add_f32
}

__global__ __launch_bounds__(1024)
void erep_edges_kernel(const int*   __restrict__ species,
                       const int*   __restrict__ esrc,
                       const int*   __restrict__ edst,
                       const float* __restrict__ dist,
                       const float* __restrict__ swt,
                       const float* __restrict__ CS,
                       const float* __restrict__ AL,
                       float*       __restrict__ out,
                       int nE) {
    // Static LDS: 207576 B < 320 KB/WGP on gfx1250; avoids dynamic-LDS opt-in.
    __shared__ float shTab[LDS_FLOATS];  // per pair: c0,a0,c1,a1,c2,a2

    const int tid = threadIdx.x;
    const int nth = blockDim.x;

    // ---- Stage CS/ALPHAS into LDS (interleaved) via async global->LDS copies ----
    for (int i = tid; i < TAB3; i += nth) {
        int p  = i / 3;
        int k  = i - 3 * p;
        int li = 6 * p + 2 * k;
#if HAVE_ASYNC_COPY
        __builtin_amdgcn_global_load_async_to_lds_b32(
            (g_i32*)(CS + i), (l_i32*)(shTab + li), 0, 0);
        __builtin_amdgcn_global_load_async_to_lds_b32(
            (g_i32*)(AL + i), (l_i32*)(shTab + li + 1), 0, 0);
#else
        shTab[li]     = CS[i];
        shTab[li + 1] = AL[i];
#endif
    }
#if HAVE_ASYNC_COPY
#if __has_builtin(__builtin_amdgcn_s_wait_asynccnt)
    __builtin_amdgcn_s_wait_asynccnt(0);
#else
    asm volatile("s_wait_asynccnt 0" ::: "memory");
#endif
#endif
    __syncthreads();

    // ---- Main loop: 4 edges / thread via b128 streaming loads ----
    const int gtid = blockIdx.x * nth + tid;
    const int gsz  = gridDim.x * nth;
    const int n4   = nE >> 2;

    const int4*   esrc4 = (const int4*)esrc;
    const int4*   edst4 = (const int4*)edst;
    const float4* dst4  = (const float4*)dist;
    const float4* swt4  = (const float4*)swt;

    for (int i = gtid; i < n4; i += gsz) {
        int4   s4 = esrc4[i];
        int4   d4 = edst4[i];
        float4 r4 = dst4[i];
        float4 w4 = swt4[i];
        do_edge(s4.x, d4.x, r4.x, w4.x, species, shTab, out);
        do_edge(s4.y, d4.y, r4.y, w4.y, species, shTab, out);
        do_edge(s4.z, d4.z, r4.z, w4.z, species, shTab, out);
        do_edge(s4.w, d4.w, r4.w, w4.w, species, shTab, out);
    }
    // Tail (nE not multiple of 4)
    for (int e = (n4 << 2) + gtid; e < nE; e += gsz) {
        do_edge(esrc[e], edst[e], dist[e], swt[e], species, shTab, out);
    }
}

__global__ void zero_out_kernel(float* __restrict__ out, int n) {
    int i = blockIdx.x * blockDim.x + threadIdx.x;
    if (i < n) out[i] = 0.0f;
}

extern "C" void kernel_launch(void* const* d_in, const int* in_sizes, int n_in,
                              void* d_out, int out_size, void* d_ws, size_t ws_size,
                              hipStream_t stream) {
    const int*   species = (const int*)  d_in[0];
    const int*   esrc    = (const int*)  d_in[1];
    const int*   edst    = (const int*)  d_in[2];
    const float* dist    = (const float*)d_in[3];
    const float* swt     = (const float*)d_in[4];
    const float* CS      = (const float*)d_in[5];
    const float* AL      = (const float*)d_in[6];
    float*       out     = (float*)d_out;

    const int nA = in_sizes[0];
    const int nE = in_sizes[1];

    // Graph replays accumulate into d_out -> must zero every launch.
    zero_out_kernel<<<(nA + 255) / 256, 256, 0, stream>>>(out, nA);

    const int threads = 1024;
    const int blocks  = 512;   // grid-stride covers 6.4M edges

    erep_edges_kernel<<<blocks, threads, 0, stream>>>(
        species, esrc, edst, dist, swt, CS, AL, out, nE);
}